// BClassifier_59760174956814
// MI455X (gfx1250) — compile-verified
//
#include <hip/hip_runtime.h>
#include <hip/hip_bf16.h>
#include <math.h>

// ---------------------------------------------------------------------------
// BClassifier forward for MI455X (gfx1250, wave32).
//  Phase 1: top-5 argmax over probs[:,1]           (latency bound, 1 WG)
//  Phase 2: p = prototype @ Wt.T + bt  [512,128]   (fp32 WMMA 16x16x4)
//           f = features[idx] @ Wt.T + bt [16,128] (fp32 WMMA, rows>=5 masked)
//  Phase 3: sim[s][c] = ||f[s]-p[c]+eps||2, normalize by row max, mean -> coding
//  Phase 4: x = relu(coding@W1.T+b1)@W2.T+b2
// ---------------------------------------------------------------------------

typedef __attribute__((ext_vector_type(2))) float v2f;
typedef __attribute__((ext_vector_type(8))) float v8f;

#define EPS 1e-6f

// ---------------- Phase 1: top-5 of probs[:,1] ------------------------------
__global__ void topk5_kernel(const float* __restrict__ probs, int N,
                             int* __restrict__ idx_out) {
    __shared__ float sval[1024];
    __shared__ int   sidx[1024];
    __shared__ int   chosen[5];
    const int t  = threadIdx.x;
    const int nt = blockDim.x;
    for (int p = 0; p < 5; ++p) {
        float bv = -1e30f;
        int   bi = -1;
        for (int i = t; i < N; i += nt) {
            bool skip = false;
            for (int q = 0; q < p; ++q) skip = skip || (chosen[q] == i);
            if (!skip) {
                float v = probs[2 * i + 1];
                if (v > bv) { bv = v; bi = i; }   // ascending i => ties keep lower index
            }
        }
        sval[t] = bv; sidx[t] = bi;
        __syncthreads();
        for (int off = nt >> 1; off > 0; off >>= 1) {
            if (t < off) {
                float v2 = sval[t + off]; int i2 = sidx[t + off];
                if (v2 > sval[t] ||
                    (v2 == sval[t] && (unsigned)i2 < (unsigned)sidx[t])) {
                    sval[t] = v2; sidx[t] = i2;
                }
            }
            __syncthreads();
        }
        if (t == 0) { chosen[p] = sidx[0]; idx_out[p] = sidx[0]; }
        __syncthreads();
    }
}

// ---------------- Phase 2: GEMM vs Wt.T via fp32 WMMA -----------------------
// MODE 0: A = prototype rows (M=512), out = p.   Pure load->WMMA inner loop.
// MODE 1: A = features[idx5[row]] for row<5 else 0 (M=16 padded), out = f5.
// One wave32 per 16x16 output tile; K advanced by 4 per V_WMMA_F32_16X16X4_F32.
template <int MODE>
__global__ void gemm_wt_kernel(const float* __restrict__ proto,
                               const float* __restrict__ features,
                               const int*   __restrict__ idx5,
                               const float* __restrict__ Wt,   // [128,512] row-major
                               const float* __restrict__ bt,   // [128]
                               float* __restrict__ out) {      // [M,128]
    const int wave   = (blockIdx.x * blockDim.x + threadIdx.x) >> 5;
    const int lane   = threadIdx.x & 31;
    const int tile_n = wave & 7;        // 128/16 = 8 tiles along N
    const int tile_m = wave >> 3;

    const int m  = lane & 15;           // A row / B col within tile
    const int kb = (lane >> 4) << 1;    // K-pair owned by this half-wave

    const int row = tile_m * 16 + m;
    const float* arow;
    float amask = 1.0f;
    if (MODE == 0) {
        arow = proto + (size_t)row * 512;
    } else {
        int r = (row < 5) ? idx5[row] : 0;   // select, not branch: EXEC stays all-1s
        amask = (row < 5) ? 1.0f : 0.0f;
        arow  = features + (size_t)r * 512;
    }
    const int n = tile_n * 16 + m;
    const float* brow = Wt + (size_t)n * 512;   // B[k][n] = Wt[n][k]

    v8f acc = {};
    for (int k = 0; k < 512; k += 4) {
        v2f a, b;
        if (MODE == 0) {
            a.x = arow[k + kb];
            a.y = arow[k + kb + 1];
        } else {
            a.x = arow[k + kb]     * amask;
            a.y = arow[k + kb + 1] * amask;
        }
        b.x = brow[k + kb];
        b.y = brow[k + kb + 1];
        acc = __builtin_amdgcn_wmma_f32_16x16x4_f32(
            /*neg_a=*/false, a, /*neg_b=*/false, b,
            /*c_mod=*/(short)0, acc, /*reuse_a=*/false, /*reuse_b=*/false);
    }

    const float bn = bt[n];
    const int mhalf = (lane >> 4) << 3;         // lanes 16-31 hold M = r+8
    for (int r = 0; r < 8; ++r) {
        int mm = tile_m * 16 + r + mhalf;
        out[(size_t)mm * 128 + n] = acc[r] + bn;
    }
}

// ---------------- Phase 3: pairwise distances -------------------------------
__global__ void dist_kernel(const float* __restrict__ f5,   // [16,128] (rows<5 valid)
                            const float* __restrict__ p,    // [512,128]
                            float* __restrict__ sim) {      // [5,512]
    const int t = blockIdx.x * blockDim.x + threadIdx.x;    // 2560 threads
    const int s = t >> 9;
    const int c = t & 511;
    const float* fr = f5 + (size_t)s * 128;
    const float* pr = p  + (size_t)c * 128;
    float acc = 0.0f;
    for (int j = 0; j < 128; ++j) {
        float d = fr[j] - pr[j] + EPS;   // torch adds eps elementwise pre-norm
        acc = fmaf(d, d, acc);
    }
    sim[t] = sqrtf(acc);
}

// ---------------- Phase 4: normalize, mean, MLP head ------------------------
__global__ void finalize_kernel(const float* __restrict__ sim,  // [5,512]
                                const float* __restrict__ W1,   // [128,512]
                                const float* __restrict__ b1,   // [128]
                                const float* __restrict__ W2,   // [2,128]
                                const float* __restrict__ b2,   // [2]
                                float* __restrict__ out) {      // [2 + 512]
    __shared__ float red[256];
    __shared__ float smax[5];
    __shared__ float coding_s[512];
    __shared__ float h_s[128];
    const int t = threadIdx.x;   // 256 threads

    for (int s = 0; s < 5; ++s) {
        float mv = -1e30f;
        for (int c = t; c < 512; c += 256) mv = fmaxf(mv, sim[s * 512 + c]);
        red[t] = mv;
        __syncthreads();
        for (int off = 128; off > 0; off >>= 1) {
            if (t < off) red[t] = fmaxf(red[t], red[t + off]);
            __syncthreads();
        }
        if (t == 0) smax[s] = red[0];
        __syncthreads();
    }

    for (int c = t; c < 512; c += 256) {
        float acc = 0.0f;
        for (int s = 0; s < 5; ++s) acc += sim[s * 512 + c] / smax[s];
        float cod = acc * 0.2f;          // mean over 5
        coding_s[c] = cod;
        out[2 + c]  = cod;               // coding output
    }
    __syncthreads();

    if (t < 128) {
        float acc = b1[t];
        for (int c = 0; c < 512; ++c) acc = fmaf(coding_s[c], W1[t * 512 + c], acc);
        h_s[t] = fmaxf(acc, 0.0f);
    }
    __syncthreads();

    if (t < 2) {
        float acc = b2[t];
        for (int j = 0; j < 128; ++j) acc = fmaf(h_s[j], W2[t * 128 + j], acc);
        out[t] = acc;                    // x output
    }
}

// ---------------------------------------------------------------------------
extern "C" void kernel_launch(void* const* d_in, const int* in_sizes, int n_in,
                              void* d_out, int out_size, void* d_ws, size_t ws_size,
                              hipStream_t stream) {
    const float* features = (const float*)d_in[0];  // [N,512]
    const float* probs    = (const float*)d_in[1];  // [N,2]
    const float* proto    = (const float*)d_in[2];  // [512,512]
    const float* Wt       = (const float*)d_in[3];  // [128,512]
    const float* bt       = (const float*)d_in[4];  // [128]
    const float* W1       = (const float*)d_in[5];  // [128,512]
    const float* b1       = (const float*)d_in[6];  // [128]
    const float* W2       = (const float*)d_in[7];  // [2,128]
    const float* b2       = (const float*)d_in[8];  // [2]
    float* out = (float*)d_out;                     // [2 + 512]

    const int N = in_sizes[1] / 2;

    // workspace layout (floats): [idx5(32 slots)][f5 16*128][p 512*128][sim 5*512]
    float* ws   = (float*)d_ws;
    int*   idx5 = (int*)d_ws;
    float* f5   = ws + 32;
    float* p    = f5 + 16 * 128;
    float* sim  = p + 512 * 128;

    topk5_kernel<<<1, 1024, 0, stream>>>(probs, N, idx5);
    // p = prototype @ Wt.T + bt : 32x8 tiles = 256 waves = 32 blocks of 8 waves
    gemm_wt_kernel<0><<<32, 256, 0, stream>>>(proto, features, idx5, Wt, bt, p);
    // f = features[idx] @ Wt.T + bt : 1x8 tiles = 8 waves = 1 block
    gemm_wt_kernel<1><<<1, 256, 0, stream>>>(proto, features, idx5, Wt, bt, f5);
    dist_kernel<<<10, 256, 0, stream>>>(f5, p, sim);
    finalize_kernel<<<1, 256, 0, stream>>>(sim, W1, b1, W2, b2, out);
}